// GCN_54984171323947
// MI455X (gfx1250) — compile-verified
//
#include <hip/hip_runtime.h>

#define NN 100000
#define NE 3200000
#define NG 1024

typedef __attribute__((ext_vector_type(2))) float v2f;
typedef __attribute__((ext_vector_type(8))) float v8f;

// ---------------------------------------------------------------- degree / norm

__global__ void k_init_deg(float* __restrict__ deg) {
  int i = blockIdx.x * blockDim.x + threadIdx.x;
  if (i < NN) deg[i] = 1.0f;  // self loop contributes 1 to every node degree
}

__global__ void k_deg_edges(const int* __restrict__ col, float* __restrict__ deg) {
  int e = blockIdx.x * blockDim.x + threadIdx.x;
  if (e < NE) {
    int c = col[e];
    if ((unsigned)c < (unsigned)NN) atomicAdd(&deg[c], 1.0f);
  }
}

__global__ void k_dinv(float* __restrict__ dd) {
  int i = blockIdx.x * blockDim.x + threadIdx.x;
  if (i < NN) {
    float d = dd[i];
    dd[i] = (d > 0.0f) ? rsqrtf(d) : 0.0f;
  }
}

// ---------------------------------------------------------------- h0 = [x[:,0]@W_emb.T + b_emb , xyz , 0pad]

__global__ void k_h0(const float* __restrict__ x, const float* __restrict__ W_emb,
                     const float* __restrict__ b_emb, float* __restrict__ h0) {
  int i = blockIdx.x * blockDim.x + threadIdx.x;
  if (i < NN) {
    const float a = x[(size_t)i * 4 + 0];
    float* o = h0 + (size_t)i * 8;
    o[0] = fmaf(a, W_emb[0], b_emb[0]);
    o[1] = fmaf(a, W_emb[1], b_emb[1]);
    o[2] = fmaf(a, W_emb[2], b_emb[2]);
    o[3] = fmaf(a, W_emb[3], b_emb[3]);
    o[4] = x[(size_t)i * 4 + 1];
    o[5] = x[(size_t)i * 4 + 2];
    o[6] = x[(size_t)i * 4 + 3];
    o[7] = 0.0f;                      // K-padding column (W row 7 is also zero)
  }
}

// ---------------------------------------------------------------- WMMA dense transforms (exact f32)
// A 16x4 f32 layout: lane L holds row M=L%16; VGPR v holds K = 2*(L/16)+v.
// B 4x16 layout:     lane L holds col N=L%16; VGPR v holds K = 2*(L/16)+v.
// D 16x16 f32:       VGPR i, lane L -> D[M = i + 8*(L/16)][N = L%16].

// t1(N x 16) = h0(N x 8) @ W1p(8 x 16), W1p[k][n] = (k<7) ? W1[n][k] : 0
__global__ void k_xform1_wmma(const float* __restrict__ h0, const float* __restrict__ W1,
                              float* __restrict__ t1, int ntiles) {
  const int wave = (int)((blockIdx.x * blockDim.x + threadIdx.x) >> 5);
  const int lane = (int)(threadIdx.x & 31u);
  if (wave >= ntiles) return;          // wave-uniform: EXEC stays all-ones for WMMA
  const int m  = lane & 15;            // A row == B col == D col
  const int kh = lane >> 4;            // K half selector

  const float* rowp = h0 + ((size_t)wave * 16 + m) * 8;
  v2f a0, a1, b0, b1;
  a0.x = rowp[2 * kh + 0];
  a0.y = rowp[2 * kh + 1];
  a1.x = rowp[2 * kh + 4];
  a1.y = rowp[2 * kh + 5];

  const float* wrow = W1 + m * 7;      // W1 is 16x7 row-major; B[k][n] = W1[n][k]
  b0.x = wrow[2 * kh + 0];
  b0.y = wrow[2 * kh + 1];
  b1.x = wrow[2 * kh + 4];
  b1.y = (2 * kh + 5 < 7) ? wrow[2 * kh + 5] : 0.0f;  // k==7 zero pad

  v8f c = {};
  c = __builtin_amdgcn_wmma_f32_16x16x4_f32(false, a0, false, b0, (short)0, c, false, false);
  c = __builtin_amdgcn_wmma_f32_16x16x4_f32(false, a1, false, b1, (short)0, c, false, false);

  float* outp = t1 + ((size_t)wave * 16 + 8 * kh) * 16 + m;
#pragma unroll
  for (int i = 0; i < 8; ++i) outp[(size_t)i * 16] = c[i];
}

// t2(N x 8) = h1(N x 16) @ W2p(16 x 16), W2p[k][n] = (n<8) ? W2[n][k] : 0
__global__ void k_xform2_wmma(const float* __restrict__ h1, const float* __restrict__ W2,
                              float* __restrict__ t2, int ntiles) {
  const int wave = (int)((blockIdx.x * blockDim.x + threadIdx.x) >> 5);
  const int lane = (int)(threadIdx.x & 31u);
  if (wave >= ntiles) return;
  const int n  = lane & 15;
  const int kh = lane >> 4;

  const float* rowp = h1 + ((size_t)wave * 16 + n) * 16;
  v8f c = {};
#pragma unroll
  for (int s = 0; s < 4; ++s) {
    v2f a, b;
    a.x = rowp[4 * s + 2 * kh + 0];
    a.y = rowp[4 * s + 2 * kh + 1];
    if (n < 8) {                       // W2 is 8x16 row-major
      b.x = W2[n * 16 + 4 * s + 2 * kh + 0];
      b.y = W2[n * 16 + 4 * s + 2 * kh + 1];
    } else {
      b.x = 0.0f;
      b.y = 0.0f;
    }
    c = __builtin_amdgcn_wmma_f32_16x16x4_f32(false, a, false, b, (short)0, c, false, false);
  }
  if (n < 8) {
    float* outp = t2 + ((size_t)wave * 16 + 8 * kh) * 8 + n;
#pragma unroll
    for (int i = 0; i < 8; ++i) outp[(size_t)i * 8] = c[i];
  }
}

// ---------------------------------------------------------------- self-loop init + edge scatter

__global__ void k_selfloop16(const float* __restrict__ t, const float* __restrict__ dinv,
                             float* __restrict__ agg) {
  int tid = blockIdx.x * blockDim.x + threadIdx.x;
  if (tid < NN * 16) {
    float di = dinv[tid >> 4];
    agg[tid] = t[tid] * di * di;
  }
}

__global__ void k_selfloop8(const float* __restrict__ t, const float* __restrict__ dinv,
                            float* __restrict__ agg) {
  int tid = blockIdx.x * blockDim.x + threadIdx.x;
  if (tid < NN * 8) {
    float di = dinv[tid >> 3];
    agg[tid] = t[tid] * di * di;
  }
}

__global__ void k_scatter16(const int* __restrict__ row, const int* __restrict__ col,
                            const float* __restrict__ dinv, const float* __restrict__ t,
                            float* __restrict__ agg) {
  int tid = blockIdx.x * blockDim.x + threadIdx.x;
  if (tid < NE * 16) {
    int e = tid >> 4, f = tid & 15;
    int r = row[e], cc = col[e];
    if ((unsigned)r < (unsigned)NN && (unsigned)cc < (unsigned)NN) {
      float nrm = dinv[r] * dinv[cc];
      atomicAdd(&agg[(size_t)cc * 16 + f], t[(size_t)r * 16 + f] * nrm);
    }
  }
}

__global__ void k_scatter8(const int* __restrict__ row, const int* __restrict__ col,
                           const float* __restrict__ dinv, const float* __restrict__ t,
                           float* __restrict__ agg) {
  int tid = blockIdx.x * blockDim.x + threadIdx.x;
  if (tid < NE * 8) {
    int e = tid >> 3, f = tid & 7;
    int r = row[e], cc = col[e];
    if ((unsigned)r < (unsigned)NN && (unsigned)cc < (unsigned)NN) {
      float nrm = dinv[r] * dinv[cc];
      atomicAdd(&agg[(size_t)cc * 8 + f], t[(size_t)r * 8 + f] * nrm);
    }
  }
}

// ---------------------------------------------------------------- bias + tanh (in place)

__global__ void k_act(float* __restrict__ a, const float* __restrict__ bias,
                      int fmask, int total) {
  int tid = blockIdx.x * blockDim.x + threadIdx.x;
  if (tid < total) a[tid] = tanhf(a[tid] + bias[tid & fmask]);
}

// ---------------------------------------------------------------- mean pool + final projection

__global__ void k_pool_init(float* __restrict__ psum, float* __restrict__ cnt) {
  int tid = blockIdx.x * blockDim.x + threadIdx.x;
  if (tid < NG * 8) psum[tid] = 0.0f;
  if (tid < NG) cnt[tid] = 0.0f;
}

__global__ void k_pool(const int* __restrict__ batch, const float* __restrict__ h2,
                       float* __restrict__ psum, float* __restrict__ cnt) {
  int tid = blockIdx.x * blockDim.x + threadIdx.x;
  if (tid < NN * 8) {
    int i = tid >> 3, f = tid & 7;
    int g = batch[i];
    if ((unsigned)g < (unsigned)NG) {
      atomicAdd(&psum[(size_t)g * 8 + f], h2[tid]);
      if (f == 0) atomicAdd(&cnt[g], 1.0f);
    }
  }
}

__global__ void k_final(const float* __restrict__ psum, const float* __restrict__ cnt,
                        const float* __restrict__ W3, const float* __restrict__ b3,
                        float* __restrict__ out) {
  int g = blockIdx.x * blockDim.x + threadIdx.x;
  if (g < NG) {
    float inv = 1.0f / fmaxf(cnt[g], 1.0f);
    float acc = b3[0];
#pragma unroll
    for (int f = 0; f < 8; ++f) acc = fmaf(psum[g * 8 + f] * inv, W3[f], acc);
    out[g] = acc;
  }
}

// ---------------------------------------------------------------- launcher

extern "C" void kernel_launch(void* const* d_in, const int* in_sizes, int n_in,
                              void* d_out, int out_size, void* d_ws, size_t ws_size,
                              hipStream_t stream) {
  (void)in_sizes; (void)n_in; (void)out_size; (void)ws_size;

  const float* x     = (const float*)d_in[0];
  const int*   ei    = (const int*)d_in[1];   // (2, E): row = ei, col = ei + NE
  const int*   batch = (const int*)d_in[2];
  const float* W_emb = (const float*)d_in[3];
  const float* b_emb = (const float*)d_in[4];
  const float* W1    = (const float*)d_in[5]; // 16 x 7
  const float* b1    = (const float*)d_in[6];
  const float* W2    = (const float*)d_in[7]; // 8 x 16
  const float* b2    = (const float*)d_in[8];
  const float* W3    = (const float*)d_in[9]; // 1 x 8
  const float* b3    = (const float*)d_in[10];
  float* out = (float*)d_out;

  const int* rowp = ei;
  const int* colp = ei + NE;

  // Workspace layout (floats), ~16.5 MB total; all regions written before read.
  float* ws   = (float*)d_ws;
  float* dinv = ws;                         // NN        (deg, then dinv in place)
  float* h0   = dinv + NN;                  // NN*8      (later reused as t2)
  float* t1   = h0 + (size_t)NN * 8;        // NN*16     (later reused as agg2/h2)
  float* agg1 = t1 + (size_t)NN * 16;       // NN*16     (h1 in place)
  float* psum = agg1 + (size_t)NN * 16;     // NG*8
  float* cnt  = psum + NG * 8;              // NG
  float* t2   = h0;                         // NN*8 reuse
  float* agg2 = t1;                         // NN*8 reuse

  const int B = 256;
  auto cdiv = [](long long a, long long b) { return (int)((a + b - 1) / b); };

  // normalization
  k_init_deg <<<cdiv(NN, B), B, 0, stream>>>(dinv);
  k_deg_edges<<<cdiv(NE, B), B, 0, stream>>>(colp, dinv);
  k_dinv     <<<cdiv(NN, B), B, 0, stream>>>(dinv);

  // embedding + concat
  k_h0<<<cdiv(NN, B), B, 0, stream>>>(x, W_emb, b_emb, h0);

  const int ntiles = NN / 16;   // 6250 (exact)
  const int wpb = B / 32;       // waves per block

  // layer 1
  k_xform1_wmma<<<cdiv(ntiles, wpb), B, 0, stream>>>(h0, W1, t1, ntiles);
  k_selfloop16 <<<cdiv((long long)NN * 16, B), B, 0, stream>>>(t1, dinv, agg1);
  k_scatter16  <<<cdiv((long long)NE * 16, B), B, 0, stream>>>(rowp, colp, dinv, t1, agg1);
  k_act        <<<cdiv((long long)NN * 16, B), B, 0, stream>>>(agg1, b1, 15, NN * 16);

  // layer 2
  k_xform2_wmma<<<cdiv(ntiles, wpb), B, 0, stream>>>(agg1, W2, t2, ntiles);
  k_selfloop8  <<<cdiv((long long)NN * 8, B), B, 0, stream>>>(t2, dinv, agg2);
  k_scatter8   <<<cdiv((long long)NE * 8, B), B, 0, stream>>>(rowp, colp, dinv, t2, agg2);
  k_act        <<<cdiv((long long)NN * 8, B), B, 0, stream>>>(agg2, b2, 7, NN * 8);

  // pooling + projection
  k_pool_init<<<cdiv(NG * 8, B), B, 0, stream>>>(psum, cnt);
  k_pool     <<<cdiv((long long)NN * 8, B), B, 0, stream>>>(batch, agg2, psum, cnt);
  k_final    <<<cdiv(NG, B), B, 0, stream>>>(psum, cnt, W3, b3, out);
}